// TFLOPLoss_40750649704811
// MI455X (gfx1250) — compile-verified
//
#include <hip/hip_runtime.h>

// ---------------- problem constants ----------------
constexpr int Bc = 32, NBc = 688, Kc = 4, Lc = 688, Sc = 688, Vc = 50;
constexpr int ROWS = Bc * Sc;          // 22016 (== Bc*NBc)
constexpr int Q4   = Sc / 4;           // 172 float4 per row (exact)
constexpr float INV_T = 10.0f;         // 1/TEMP
constexpr float NEG_BIG = -3.0e38f;

// workspace offsets (floats)
constexpr int OFF_TAGP  = 0;
constexpr int OFF_TAGV  = OFF_TAGP + ROWS;
constexpr int OFF_PTRP  = OFF_TAGV + ROWS;
constexpr int OFF_PTRH  = OFF_PTRP + ROWS;
constexpr int OFF_RB    = OFF_PTRH + ROWS;
constexpr int OFF_RV    = OFF_RB + ROWS;
constexpr int OFF_CB    = OFF_RV + ROWS;
constexpr int OFF_CV    = OFF_CB + ROWS;
constexpr int OFF_EMPTY = OFF_CV + ROWS;   // 32 floats

typedef __attribute__((ext_vector_type(2))) float v2f;
typedef __attribute__((ext_vector_type(4))) float v4f;
typedef __attribute__((ext_vector_type(8))) float v8f;

// ---------------- wave32 reductions ----------------
__device__ __forceinline__ float waveSum(float v) {
    v += __shfl_xor(v, 16, 32);
    v += __shfl_xor(v, 8, 32);
    v += __shfl_xor(v, 4, 32);
    v += __shfl_xor(v, 2, 32);
    v += __shfl_xor(v, 1, 32);
    return v;
}
__device__ __forceinline__ float waveMax(float v) {
    v = fmaxf(v, __shfl_xor(v, 16, 32));
    v = fmaxf(v, __shfl_xor(v, 8, 32));
    v = fmaxf(v, __shfl_xor(v, 4, 32));
    v = fmaxf(v, __shfl_xor(v, 2, 32));
    v = fmaxf(v, __shfl_xor(v, 1, 32));
    return v;
}
__device__ __forceinline__ float vmax4(v4f v) {
    return fmaxf(fmaxf(v.x, v.y), fmaxf(v.z, v.w));
}

// ---------------- kernel 1: tag loss per row ----------------
// rows = B*S, V=50. 8 waves/block, 1 wave per row.
__global__ void tag_loss_rows(const float* __restrict__ logits,
                              const int* __restrict__ targets,
                              float* __restrict__ perOut,
                              float* __restrict__ valOut) {
    const int lane = threadIdx.x & 31;
    const int row = blockIdx.x * 8 + (threadIdx.x >> 5);
    const float* xr = logits + (size_t)row * Vc;

    float x0 = __builtin_nontemporal_load(xr + lane);
    float x1 = (lane < Vc - 32) ? __builtin_nontemporal_load(xr + lane + 32) : NEG_BIG;
    float m = waveMax(fmaxf(x0, x1));
    float se = expf(x0 - m) + ((lane < Vc - 32) ? expf(x1 - m) : 0.0f);
    se = waveSum(se);
    float sx = x0 + ((lane < Vc - 32) ? x1 : 0.0f);
    sx = waveSum(sx);
    float logZ = m + logf(se);

    int t = targets[row];
    int ts = (t >= 0 && t < Vc) ? t : 0;
    float xt = xr[ts];                       // uniform address within wave
    float nll = logZ - xt;
    float smooth = logZ - sx * (1.0f / Vc);
    float per = 0.9f * nll + 0.1f * smooth;  // (1-SMOOTH)*nll + SMOOTH*smooth
    float valid = (t != 0) ? 1.0f : 0.0f;    // PAD == 0
    if (lane == 0) {
        perOut[row] = per * valid;
        valOut[row] = valid;
    }
}

// ---------------- kernel 2: pointer loss per row ----------------
// 1 wave per row of S=688 floats, cached as 6 x float4 per lane (b128 NT loads).
__global__ void pointer_loss_rows(const float* __restrict__ plog,
                                  const int* __restrict__ box,
                                  const unsigned char* __restrict__ dataMask,
                                  float* __restrict__ perOut,
                                  float* __restrict__ hasOut) {
    const int lane = threadIdx.x & 31;
    const int row = blockIdx.x * 8 + (threadIdx.x >> 5);
    const int b = row / NBc;
    const float* xr = plog + (size_t)row * Sc;
    const v4f* xr4 = (const v4f*)xr;

    v4f xs[6];
    float mloc = NEG_BIG;
#pragma unroll
    for (int i = 0; i < 6; ++i) {
        int q = i * 32 + lane;                   // Q4 = 172 = 5*32 + 12
        if (q < Q4) xs[i] = __builtin_nontemporal_load(xr4 + q);
        else        xs[i] = (v4f){NEG_BIG, NEG_BIG, NEG_BIG, NEG_BIG};
        mloc = fmaxf(mloc, vmax4(xs[i]));
    }
    float m = waveMax(mloc);
    float zl = 0.0f;
#pragma unroll
    for (int i = 0; i < 6; ++i) {                // sentinel -> exp() == 0
        zl += expf((xs[i].x - m) * INV_T) + expf((xs[i].y - m) * INV_T)
            + expf((xs[i].z - m) * INV_T) + expf((xs[i].w - m) * INV_T);
    }
    float Z = waveSum(zl);
    float logZ = logf(Z);

    float g = 0.0f, cnt = 0.0f;
    if (lane < Kc) {
        int idx = box[(size_t)row * Kc + lane];
        if (idx >= 0) {
            cnt = 1.0f;
            float xv = xr[idx];
            float lp = (xv - m) * INV_T - logZ;
            bool vp = dataMask[(size_t)b * (Lc + Sc) + Lc + idx] != 0;
            g = vp ? lp : -__builtin_inff();
        }
    }
    float gs = waveSum(g);
    float cs = waveSum(cnt);
    bool has = cs > 0.5f;
    float per = -(gs / fmaxf(cs, 1.0f));
    if (lane == 0) {
        perOut[row] = has ? per : 0.0f;
        hasOut[row] = has ? 1.0f : 0.0f;
    }
}

// ---------------- kernel 3: empty-pointer BCE per batch ----------------
__global__ void empty_loss_rows(const float* __restrict__ elog,
                                const unsigned char* __restrict__ dataMask,
                                const unsigned char* __restrict__ emptyMask,
                                float* __restrict__ outB) {
    const int lane = threadIdx.x & 31;
    const int b = blockIdx.x * 8 + (threadIdx.x >> 5);
    const float* xr = elog + (size_t)b * Sc;
    const unsigned char* dm = dataMask + (size_t)b * (Lc + Sc) + Lc;
    const unsigned char* em = emptyMask + (size_t)b * (Lc + Sc) + Lc;

    float ms = 0.0f, cs = 0.0f;
    for (int j = lane; j < Sc; j += 32) {
        float x = xr[j];
        bool d = dm[j] != 0, e = em[j] != 0;
        if (d || e) {
            float t = e ? 1.0f : 0.0f;
            float bce = fmaxf(x, 0.0f) - x * t + log1pf(expf(-fabsf(x)));
            ms += bce;
            cs += 1.0f;
        }
    }
    ms = waveSum(ms);
    cs = waveSum(cs);
    if (lane == 0) outB[b] = ms / (cs + 1e-08f);
}

// ---------------- kernel 4: contrastive per row (y: 0=row, 1=col) ----------------
// 1 wave per row; sim and coef rows cached as 6 x float4 per lane (b128 NT loads).
__global__ void contrastive_rows(const float* __restrict__ rsim,
                                 const float* __restrict__ rcoef,
                                 const float* __restrict__ csim,
                                 const float* __restrict__ ccoef,
                                 float* __restrict__ rb, float* __restrict__ rv,
                                 float* __restrict__ cb, float* __restrict__ cv_) {
    const int lane = threadIdx.x & 31;
    const int row = blockIdx.x * 8 + (threadIdx.x >> 5);
    const bool isRow = (blockIdx.y == 0);
    const v4f* sr4 = (const v4f*)((isRow ? rsim : csim) + (size_t)row * NBc);
    const v4f* cr4 = (const v4f*)((isRow ? rcoef : ccoef) + (size_t)row * NBc);
    float* boxOut = isRow ? rb : cb;
    float* vbOut  = isRow ? rv : cv_;

    v4f sv[6], cw[6];
    float mloc = NEG_BIG;
#pragma unroll
    for (int i = 0; i < 6; ++i) {
        int q = i * 32 + lane;
        if (q < Q4) {
            sv[i] = __builtin_nontemporal_load(sr4 + q);
            cw[i] = __builtin_nontemporal_load(cr4 + q);
        } else {
            sv[i] = (v4f){NEG_BIG, NEG_BIG, NEG_BIG, NEG_BIG};
            cw[i] = (v4f){0.0f, 0.0f, 0.0f, 0.0f};
        }
        mloc = fmaxf(mloc, vmax4(sv[i]));
    }
    float m = waveMax(mloc);
    float zl = 0.0f;
#pragma unroll
    for (int i = 0; i < 6; ++i) {                // sentinel -> exp() == 0
        zl += expf((sv[i].x - m) * INV_T) + expf((sv[i].y - m) * INV_T)
            + expf((sv[i].z - m) * INV_T) + expf((sv[i].w - m) * INV_T);
    }
    float Z = waveSum(zl);

    float wl = 0.0f, cl = 0.0f, al = 0.0f;
#pragma unroll
    for (int i = 0; i < 6; ++i) {
        bool ok = (i * 32 + lane) < Q4;
        if (ok) {
#pragma unroll
            for (int k = 0; k < 4; ++k) {
                float c = cw[i][k];
                float p = expf((sv[i][k] - m) * INV_T) / Z;
                float lp = logf(p + 1e-08f);
                if (c >= 0.0f) { wl += c * lp; al = 1.0f; }
                cl += c;
            }
        }
    }
    float wsum = waveSum(wl);
    float csum = waveSum(cl);
    float any  = waveSum(al);
    bool vb = any > 0.5f;
    float box = -(wsum / (csum + 1e-08f));
    if (lane == 0) {
        boxOut[row] = vb ? box : 0.0f;
        vbOut[row]  = vb ? 1.0f : 0.0f;
    }
}

// ---------------- WMMA f32 16x16x4 column-sum reduction ----------------
// A = ones(16x4) -> D[m][n] = sum_k B[k][n]; chain C across 4x16 tiles.
// n must be a multiple of 64 here (22016 = 344*64).
__device__ __forceinline__ float wmma_sum64(const float* __restrict__ data,
                                            int n, int lane) {
    v2f a; a.x = 1.0f; a.y = 1.0f;
    v8f c = {};
    const int off0 = ((lane & 16) << 1) + (lane & 15);  // lane>=16 -> +32
    for (int base = 0; base + 64 <= n; base += 64) {
        v2f b;
        b.x = data[base + off0];
        b.y = data[base + off0 + 16];
        c = __builtin_amdgcn_wmma_f32_16x16x4_f32(false, a, false, b,
                                                  (short)0, c, false, false);
    }
    float v = c[0];                    // both 16-lane halves hold the column sums
    v += __shfl_xor(v, 1, 32);
    v += __shfl_xor(v, 2, 32);
    v += __shfl_xor(v, 4, 32);
    v += __shfl_xor(v, 8, 32);
    return __shfl(v, 0, 32);
}

// ---------------- kernel 5: finalize (1 wave) ----------------
__global__ void finalize_losses(const float* __restrict__ ws, float* __restrict__ out) {
    const int lane = threadIdx.x;
    float tagP = wmma_sum64(ws + OFF_TAGP, ROWS, lane);
    float tagV = wmma_sum64(ws + OFF_TAGV, ROWS, lane);
    float ptrP = wmma_sum64(ws + OFF_PTRP, ROWS, lane);
    float ptrH = wmma_sum64(ws + OFF_PTRH, ROWS, lane);
    float rB   = wmma_sum64(ws + OFF_RB,   ROWS, lane);
    float rV   = wmma_sum64(ws + OFF_RV,   ROWS, lane);
    float cB   = wmma_sum64(ws + OFF_CB,   ROWS, lane);
    float cV   = wmma_sum64(ws + OFF_CV,   ROWS, lane);
    float eS   = waveSum(ws[OFF_EMPTY + lane]);   // 32 per-batch values

    float cls   = tagP / tagV;
    float ptr   = ptrP / (ptrH + 1e-06f);
    float empty = eS * (1.0f / Bc);
    float row   = rB / (rV + 1e-08f);
    float col   = cB / (cV + 1e-08f);
    float total = cls + ptr + empty + 0.5f * row + 0.5f * col;
    if (lane == 0) {
        out[0] = total; out[1] = cls; out[2] = ptr;
        out[3] = empty; out[4] = row; out[5] = col;
    }
}

// ---------------- host launcher ----------------
extern "C" void kernel_launch(void* const* d_in, const int* in_sizes, int n_in,
                              void* d_out, int out_size, void* d_ws, size_t ws_size,
                              hipStream_t stream) {
    (void)in_sizes; (void)n_in; (void)out_size; (void)ws_size;
    const float* tag_logits   = (const float*)d_in[0];
    const int*   tag_targets  = (const int*)d_in[1];
    const float* ptr_logits   = (const float*)d_in[2];
    const int*   box_indices  = (const int*)d_in[3];
    const unsigned char* data_mask  = (const unsigned char*)d_in[4];
    const unsigned char* empty_mask = (const unsigned char*)d_in[5];
    const float* empty_logits = (const float*)d_in[6];
    const float* row_sim      = (const float*)d_in[7];
    const float* col_sim      = (const float*)d_in[8];
    const float* row_coef     = (const float*)d_in[9];
    const float* col_coef     = (const float*)d_in[10];
    float* ws  = (float*)d_ws;
    float* out = (float*)d_out;

    const int blocks = ROWS / 8;  // 2752, 8 waves per 256-thread block

    tag_loss_rows<<<blocks, 256, 0, stream>>>(tag_logits, tag_targets,
                                              ws + OFF_TAGP, ws + OFF_TAGV);
    pointer_loss_rows<<<blocks, 256, 0, stream>>>(ptr_logits, box_indices, data_mask,
                                                  ws + OFF_PTRP, ws + OFF_PTRH);
    empty_loss_rows<<<Bc / 8, 256, 0, stream>>>(empty_logits, data_mask, empty_mask,
                                                ws + OFF_EMPTY);
    dim3 cgrid(blocks, 2);
    contrastive_rows<<<cgrid, 256, 0, stream>>>(row_sim, row_coef, col_sim, col_coef,
                                                ws + OFF_RB, ws + OFF_RV,
                                                ws + OFF_CB, ws + OFF_CV);
    finalize_losses<<<1, 32, 0, stream>>>(ws, out);
}